// LUFSNormalization_76416058131090
// MI455X (gfx1250) — compile-verified
//
#include <hip/hip_runtime.h>
#include <math.h>

#define ALPHA   0.99f
#define ROWS    16
#define NSAMP   2097152              // 2^21 samples per row
#define SEG     64                   // segments per row
#define SEGLEN  (NSAMP / SEG)        // 32768
#define TPB     256                  // 8 waves of 32 (wave32)
#define TILE    2048                 // floats per block tile (256 per wave)
#define NTILES  (SEGLEN / TILE)      // 16
#define NWAVES  (TPB / 32)

typedef __attribute__((ext_vector_type(2))) float v2f;
typedef __attribute__((ext_vector_type(8))) float v8f;

// ---------------------------------------------------------------------------
// CDNA5 async global->LDS copy (gfx1250), tracked by ASYNCcnt. INST_OFFSET is
// added to BOTH the LDS and global addresses (CDNA5 ISA):
// one per-lane base + two immediates covers the 8 KiB tile.
// ---------------------------------------------------------------------------
__device__ __forceinline__ void async_tile_load(const float* gbase, float* lds_base, int tid) {
  unsigned laddr = (unsigned)(unsigned long long)(void*)lds_base + (unsigned)(tid * 16);
  unsigned long long gaddr = (unsigned long long)(const void*)(gbase + tid * 4);
  asm volatile("global_load_async_to_lds_b128 %0, %1, off"             :: "v"(laddr), "v"(gaddr) : "memory");
  asm volatile("global_load_async_to_lds_b128 %0, %1, off offset:4096" :: "v"(laddr), "v"(gaddr) : "memory");
}

__device__ __forceinline__ void wait_async() {
  asm volatile("s_wait_asynccnt 0" ::: "memory");
}

__device__ __forceinline__ float powN(float b, int n) {
  float r = 1.0f;
  for (int i = 0; i < n; ++i) r *= b;
  return r;
}
__device__ __forceinline__ float tcoef(int n) {   // Toeplitz entry: a^n (n>=0), else 0
  return (n < 0) ? 0.0f : powN(ALPHA, n);
}

// ---------------------------------------------------------------------------
// Segment kernel. Per wave: 16 chunks x 16 samples. The dense prefix operator
// p = T*b is computed on the matrix pipe as 4 chained V_WMMA_F32_16X16X4_F32;
// VALU handles only the alpha^16-carry scan over chunks, the homogeneous term
// and (SUMSQ) the squared accumulation. Maps (d,y): v -> d*v + y.
// SUMSQ=false: emit b_seg (segment output with zero incoming state).
// SUMSQ=true : consume true incoming state y_in, emit sum of y^2.
// ---------------------------------------------------------------------------
template <bool SUMSQ>
__global__ __launch_bounds__(TPB) void seg_scan_kernel(const float* __restrict__ audio,
                                                       float* __restrict__ b_seg,
                                                       const float* __restrict__ y_in,
                                                       float* __restrict__ ss_out) {
  __shared__ float buf[2][TILE];                 // 16 KiB double buffer
  __shared__ float s_wd[NWAVES], s_wy[NWAVES];
  __shared__ float s_carry;
  __shared__ float s_red[NWAVES];

  const int seg  = blockIdx.x;
  const int row  = blockIdx.y;
  const int tid  = threadIdx.x;
  const int lane = tid & 31;
  const int wave = tid >> 5;
  const int c    = lane & 15;                    // column / chunk index
  const int half = lane >> 4;                    // 0: rows 0-7, 1: rows 8-15

  const long long seg_base = (long long)row * NSAMP + (long long)seg * SEGLEN;
  const float* gseg = audio + seg_base;

  // Hoisted constants.
  const float D16  = powN(ALPHA, 16);            // per-chunk homogeneous decay
  const float D256 = powN(D16, 16);              // per-wave (16 chunks) decay
  const float dcp  = powN(D16, c);               // D16^c for chunk-incoming term

  // A-matrix (T chunks) per WMMA g: lane supplies T[c][4g+2*half+{0,1}].
  v2f Amat[4];
#pragma unroll
  for (int g = 0; g < 4; ++g) {
    int k0 = 4 * g + 2 * half;
    Amat[g].x = tcoef(c - k0);
    Amat[g].y = tcoef(c - (k0 + 1));
  }
  // alpha^(row+1) for this lane's 8 rows (row = half*8 + v).
  float rp[8];
  rp[0] = powN(ALPHA, half * 8 + 1);
#pragma unroll
  for (int v = 1; v < 8; ++v) rp[v] = rp[v - 1] * ALPHA;

  // Incoming x for the segment; synthetic value at the row start makes the
  // uniform recurrence reproduce y[0] = x[0].
  float x_carry;
  if (seg == 0) {
    float x0 = audio[(long long)row * NSAMP];
    x_carry = x0 * (1.0f - 1.0f / ALPHA);
  } else {
    x_carry = gseg[-1];
  }
  float y_carry = SUMSQ ? y_in[row * SEG + seg] : 0.0f;
  float ss = 0.0f;

  async_tile_load(gseg, buf[0], tid);

  for (int k = 0; k < NTILES; ++k) {
    const int p = k & 1;
    wait_async();
    __syncthreads();                             // tile k resident in buf[p]

    float xlast = buf[p][TILE - 1];              // x carry for next tile

    if (k + 1 < NTILES)                          // prefetch next tile
      async_tile_load(gseg + (long long)(k + 1) * TILE, buf[1 - p], tid);

    // --- matrix-pipe prefix: D = T * B, chained over K ---
    v8f acc = {};
#pragma unroll
    for (int g = 0; g < 4; ++g) {
      int e = wave * 256 + c * 16 + 4 * g + 2 * half;   // sample index in tile
      float xa = (e == 0) ? x_carry : buf[p][e - 1];    // only tid==0 hits e==0
      float xb = buf[p][e];
      float xc = buf[p][e + 1];
      v2f bv;
      bv.x = ALPHA * (xb - xa);                  // b[e]
      bv.y = ALPHA * (xc - xb);                  // b[e+1]
      acc = __builtin_amdgcn_wmma_f32_16x16x4_f32(
          false, Amat[g], false, bv, (short)0, acc, false, false);
    }

    // --- chunk-carry scan within the wave (factor D16 over 16 chunk ends) ---
    float e7 = acc[7];                           // lane 16+c holds e_c = D[15][c]
    float f = 0.0f, ft = 0.0f;
#pragma unroll
    for (int i = 0; i < 16; ++i) {
      float ei = __shfl(e7, 16 + i, 32);
      ft = D16 * ft + ei;                        // full wave fold
      if (i < c) f = D16 * f + ei;               // exclusive prefix for chunk c
    }
    if (lane == 0) { s_wd[wave] = D256; s_wy[wave] = ft; }
    __syncthreads();

    // Wave-exclusive prefix over wave maps (in order; deterministic).
    float ad = 1.0f, ay = 0.0f;
    for (int i = 0; i < wave; ++i) {
      float wd = s_wd[i], wy = s_wy[i];
      ay = wd * ay + wy;
      ad = ad * wd;
    }
    float y_wave_in = ad * y_carry + ay;
    float inc = dcp * y_wave_in + f;             // incoming state of chunk c

    if (SUMSQ) {
#pragma unroll
      for (int v = 0; v < 8; ++v) {
        float yv = acc[v] + rp[v] * inc;         // exact filtered sample
        ss += yv * yv;
      }
    }

    // Tile carry-out = full-tile map applied to incoming carry.
    if (tid == 0) {
      float bd = 1.0f, by = 0.0f;
#pragma unroll
      for (int i = 0; i < NWAVES; ++i) {
        float wd = s_wd[i], wy = s_wy[i];
        by = wd * by + wy;
        bd = bd * wd;
      }
      s_carry = bd * y_carry + by;
    }
    __syncthreads();
    y_carry = s_carry;
    x_carry = xlast;
  }

  if (SUMSQ) {
#pragma unroll
    for (int off = 16; off > 0; off >>= 1) ss += __shfl_down(ss, off, 32);
    if (lane == 0) s_red[wave] = ss;
    __syncthreads();
    if (tid == 0) {
      float t = 0.0f;
#pragma unroll
      for (int i = 0; i < NWAVES; ++i) t += s_red[i];
      ss_out[row * SEG + seg] = t;
    }
  } else {
    if (tid == 0) b_seg[row * SEG + seg] = y_carry;
  }
}

// Serial per-row combine of segment maps -> incoming state per segment.
__global__ void seg_combine_kernel(const float* __restrict__ b_seg,
                                   float* __restrict__ y_in) {
  int row = threadIdx.x;
  if (row >= ROWS) return;
  float A = powf(ALPHA, (float)SEGLEN);          // underflows to ~0; kept exact
  float y = 0.0f;
  for (int s = 0; s < SEG; ++s) {
    y_in[row * SEG + s] = y;
    y = A * y + b_seg[row * SEG + s];
  }
}

// Per-row: mean square -> LUFS -> clipped gain (dB) -> linear gain.
__global__ void gain_kernel(const float* __restrict__ ss,
                            float* __restrict__ gain) {
  int row = threadIdx.x;
  if (row >= ROWS) return;
  float sum = 0.0f;
  for (int s = 0; s < SEG; ++s) sum += ss[row * SEG + s];
  float ms   = sum / (float)NSAMP;
  float lufs = -0.691f + 10.0f * log10f(ms + 1e-8f);
  float gdb  = fminf(fmaxf(-23.0f - lufs, -30.0f), 30.0f);
  gain[row]  = expf(gdb * 0.11512925465f);       // 10^(gdb/20)
}

// out = audio * gain[row], float4 vectorized (row length divisible by 4).
__global__ __launch_bounds__(256) void scale_kernel(const float* __restrict__ audio,
                                                    const float* __restrict__ gain,
                                                    float* __restrict__ out) {
  long long i = (long long)blockIdx.x * blockDim.x + threadIdx.x;  // float4 idx
  int row = (int)(i >> 19);                      // NSAMP/4 = 2^19 float4 per row
  float g = gain[row];
  const float4* a4 = (const float4*)audio;
  float4* o4 = (float4*)out;
  float4 v = a4[i];
  v.x *= g; v.y *= g; v.z *= g; v.w *= g;
  o4[i] = v;
}

extern "C" void kernel_launch(void* const* d_in, const int* in_sizes, int n_in,
                              void* d_out, int out_size, void* d_ws, size_t ws_size,
                              hipStream_t stream) {
  (void)in_sizes; (void)n_in; (void)out_size; (void)ws_size;
  const float* audio = (const float*)d_in[0];
  float* out  = (float*)d_out;
  float* ws   = (float*)d_ws;
  float* b_seg = ws;                             // ROWS*SEG
  float* y_in  = ws + 1 * ROWS * SEG;            // ROWS*SEG
  float* ssv   = ws + 2 * ROWS * SEG;            // ROWS*SEG
  float* gain  = ws + 3 * ROWS * SEG;            // ROWS

  dim3 grid(SEG, ROWS);
  seg_scan_kernel<false><<<grid, TPB, 0, stream>>>(audio, b_seg, nullptr, nullptr);
  seg_combine_kernel<<<1, 32, 0, stream>>>(b_seg, y_in);
  seg_scan_kernel<true><<<grid, TPB, 0, stream>>>(audio, nullptr, y_in, ssv);
  gain_kernel<<<1, 32, 0, stream>>>(ssv, gain);
  scale_kernel<<<(ROWS * (NSAMP / 4)) / 256, 256, 0, stream>>>(audio, gain, out);
}